// GCN_pia1_44306882625586
// MI455X (gfx1250) — compile-verified
//
#include <hip/hip_runtime.h>
#include <hip/hip_bf16.h>
#include <math.h>

// GCN layer on MI455X (gfx1250):
//   support = x @ W                (16384x512 @ 512x256)
//   embed1  = adj @ support + b    (16384x16384 @ 16384x256)  <- 137 GFLOP, adj = 1 GiB streamed once
//   logp    = log_softmax(embed1, axis=1)  (fused epilogue)
//
// fp32 WMMA (V_WMMA_F32_16X16X4_F32) for exact fp32 FMA semantics. Both A and B
// K-slabs are double-buffered in LDS and filled with gfx1250 async-to-LDS copies
// (ASYNCcnt) issued one tile ahead; B is stored pair-interleaved so each 4x16
// B fragment is one ds_load_b64.

typedef __attribute__((ext_vector_type(2))) float v2f;
typedef __attribute__((ext_vector_type(8))) float v8f;

#define NROWS 16384
#define FHID  256
#define KT    16               // K-tile depth staged in LDS
#define BROWS 32               // output rows per block
#define BLOCK_THREADS 512      // 16 wave32 waves (4 per SIMD32)
#define ASTRIDE 20             // A tile row stride (floats): 16B aligned, conflict-free
#define BSTRIDE 520            // B tile k-pair stride (floats): FHID*2 + 8 pad

#if __has_builtin(__builtin_amdgcn_global_load_async_to_lds_b128) && \
    __has_builtin(__builtin_amdgcn_global_load_async_to_lds_b32)
#define USE_ASYNC_LDS 1
#else
#define USE_ASYNC_LDS 0
#endif

// Exact parameter types per hipcc diagnostic:
//   b128: (int4_gcc __device__ *, int4_gcc __shared__ *, Imm, Imm)
//   b32 : (int      __device__ *, int      __shared__ *, Imm, Imm)
typedef int gcc_int4 __attribute__((vector_size(4 * sizeof(int))));
typedef __attribute__((address_space(1))) gcc_int4 as1_int4;
typedef __attribute__((address_space(3))) gcc_int4 as3_int4;
typedef __attribute__((address_space(1))) int      as1_int;
typedef __attribute__((address_space(3))) int      as3_int;

__device__ __forceinline__ unsigned long long gbits(const void* p) {
    return (unsigned long long)p;            // generic global ptr == AS1 bits
}
__device__ __forceinline__ unsigned lbits(const void* p) {
    return (unsigned)(unsigned long long)p;  // low 32 bits = LDS byte offset
}

__device__ __forceinline__ void copy_b128(const float* g, float* l) {
#if USE_ASYNC_LDS
    __builtin_amdgcn_global_load_async_to_lds_b128(
        (as1_int4*)gbits(g), (as3_int4*)lbits(l), 0, 0);
#else
    *reinterpret_cast<float4*>(l) = *reinterpret_cast<const float4*>(g);
#endif
}
__device__ __forceinline__ void copy_b32(const float* g, float* l) {
#if USE_ASYNC_LDS
    __builtin_amdgcn_global_load_async_to_lds_b32(
        (as1_int*)gbits(g), (as3_int*)lbits(l), 0, 0);
#else
    *l = *g;
#endif
}
__device__ __forceinline__ void wait_async() {
#if USE_ASYNC_LDS
#if __has_builtin(__builtin_amdgcn_s_wait_asynccnt)
    __builtin_amdgcn_s_wait_asynccnt(0);
#else
    asm volatile("s_wait_asynccnt 0x0" ::: "memory");
#endif
#endif
}

__global__ __launch_bounds__(BLOCK_THREADS)
void gcn_gemm_strip(const float* __restrict__ A,      // M x K, row-major
                    const float* __restrict__ B,      // K x 256, row-major
                    const float* __restrict__ bias,   // 256 or nullptr
                    float* __restrict__ out_main,     // M x 256 (support or embed1)
                    float* __restrict__ out_logp,     // M x 256 or nullptr
                    int K)
{
    __shared__ float tileA[2][BROWS * ASTRIDE];       // 2 x 2.5 KB
    __shared__ float tileB[2][(KT / 2) * BSTRIDE];    // 2 x 16.25 KB, pair-interleaved

    const int tid  = threadIdx.x;
    const int wave = tid >> 5;          // 0..15
    const int lane = tid & 31;
    const int hi   = lane >> 4;         // selects K pair {0,1} vs {2,3}
    const int l15  = lane & 15;

    const int rowHalf  = wave >> 3;             // two 16-row groups
    const int colGroup = wave & 7;              // eight 32-col groups
    const int rowBase  = blockIdx.x * BROWS;
    const int aRow     = rowHalf * 16 + l15;    // block-local A-fragment row
    const int col0     = colGroup * 32 + l15;
    const int col1     = col0 + 16;

    v8f c0 = {0.f,0.f,0.f,0.f,0.f,0.f,0.f,0.f};
    v8f c1 = c0;

    // Issue all copies for one K-tile (async path: non-blocking, ASYNCcnt++)
    auto load_tiles = [&](int buf, int kt) {
        if (tid < BROWS * (KT / 4)) {           // 128 x b128 chunks of A
            const int row = tid >> 2;
            const int c   = (tid & 3) << 2;
            copy_b128(&A[(size_t)(rowBase + row) * K + kt + c],
                      &tileA[buf][row * ASTRIDE + c]);
        }
        #pragma unroll
        for (int q = 0; q < (KT * FHID) / BLOCK_THREADS; ++q) {   // 8 x b32, scattered
            const int e = tid + q * BLOCK_THREADS;                // 0..4095
            const int k = e >> 8;
            const int n = e & 255;
            copy_b32(&B[(size_t)(kt + k) * FHID + n],
                     &tileB[buf][(k >> 1) * BSTRIDE + n * 2 + (k & 1)]);
        }
    };

    int buf = 0;
    load_tiles(0, 0);
    wait_async();
    __syncthreads();

    for (int kt = 0; kt < K; kt += KT) {
        if (kt + KT < K) load_tiles(buf ^ 1, kt + KT);   // prefetch next tile (async)

        #pragma unroll
        for (int kk = 0; kk < KT; kk += 4) {
            const int kg = kk + 2 * hi;                  // even
            // A fragment (16x4): row = lane&15, two consecutive K values
            const float2 av = *reinterpret_cast<const float2*>(
                &tileA[buf][aRow * ASTRIDE + kg]);
            // B fragment (4x16): pair-interleaved -> {B[kg][col], B[kg+1][col]} contiguous
            const float2 b0v = *reinterpret_cast<const float2*>(
                &tileB[buf][(kg >> 1) * BSTRIDE + col0 * 2]);
            const float2 b1v = *reinterpret_cast<const float2*>(
                &tileB[buf][(kg >> 1) * BSTRIDE + col1 * 2]);
            v2f a  = {av.x,  av.y};
            v2f b0 = {b0v.x, b0v.y};
            v2f b1 = {b1v.x, b1v.y};
            c0 = __builtin_amdgcn_wmma_f32_16x16x4_f32(false, a, false, b0,
                                                       (short)0, c0, false, false);
            c1 = __builtin_amdgcn_wmma_f32_16x16x4_f32(false, a, false, b1,
                                                       (short)0, c1, false, false);
        }

        wait_async();       // next tile landed (own issues) ...
        __syncthreads();    // ... and everyone is done with the current buffer
        buf ^= 1;
    }

    const float bv0 = bias ? bias[col0] : 0.f;
    const float bv1 = bias ? bias[col1] : 0.f;

    if (out_logp == nullptr) {
        // GEMM1 epilogue: C/D layout -> VGPR i holds rows i (lo lanes) / i+8 (hi lanes)
        #pragma unroll
        for (int i = 0; i < 8; ++i) {
            const size_t r = (size_t)(rowBase + rowHalf * 16 + i + 8 * hi);
            out_main[r * FHID + col0] = c0[i] + bv0;
            out_main[r * FHID + col1] = c1[i] + bv1;
        }
        return;
    }

    // GEMM2 epilogue: stage 32x256 strip in LDS (tileB is free after last barrier),
    // then per-row log_softmax with wave32 shuffles. Wave w reduces rows 2w, 2w+1.
    float* stage = (float*)tileB;   // 8320 floats >= 32*256
    #pragma unroll
    for (int i = 0; i < 8; ++i) {
        const int r = rowHalf * 16 + i + 8 * hi;
        stage[r * FHID + col0] = c0[i] + bv0;
        stage[r * FHID + col1] = c1[i] + bv1;
    }
    __syncthreads();

    #pragma unroll
    for (int rr = 0; rr < 2; ++rr) {
        const int r  = wave * 2 + rr;
        const size_t rg = (size_t)(rowBase + r);

        float v[8];
        float m = -__builtin_inff();
        #pragma unroll
        for (int j = 0; j < 8; ++j) {
            v[j] = stage[r * FHID + lane + 32 * j];
            m = fmaxf(m, v[j]);
        }
        #pragma unroll
        for (int off = 16; off > 0; off >>= 1)
            m = fmaxf(m, __shfl_xor(m, off, 32));

        float s = 0.f;
        #pragma unroll
        for (int j = 0; j < 8; ++j) s += expf(v[j] - m);
        #pragma unroll
        for (int off = 16; off > 0; off >>= 1)
            s += __shfl_xor(s, off, 32);

        const float logZ = m + logf(s);
        #pragma unroll
        for (int j = 0; j < 8; ++j) {
            const size_t idx = rg * FHID + lane + 32 * j;
            out_logp[idx] = v[j] - logZ;   // logp
            out_main[idx] = v[j];          // embed1
        }
    }
}

extern "C" void kernel_launch(void* const* d_in, const int* in_sizes, int n_in,
                              void* d_out, int out_size, void* d_ws, size_t ws_size,
                              hipStream_t stream) {
    (void)in_sizes; (void)n_in; (void)out_size; (void)ws_size;

    const float* x   = (const float*)d_in[0];   // 16384 x 512
    const float* adj = (const float*)d_in[1];   // 16384 x 16384
    const float* W   = (const float*)d_in[2];   // 512 x 256
    const float* b   = (const float*)d_in[3];   // 256

    float* logp    = (float*)d_out;                          // output 0
    float* embed1  = (float*)d_out + (size_t)NROWS * FHID;   // output 1
    float* support = (float*)d_ws;                           // 16 MB scratch

    dim3 grid(NROWS / BROWS);   // 512 blocks

    // support = x @ W
    gcn_gemm_strip<<<grid, BLOCK_THREADS, 0, stream>>>(
        x, W, nullptr, support, nullptr, 512);

    // embed1 = adj @ support + b ; logp = log_softmax(embed1) (fused)
    gcn_gemm_strip<<<grid, BLOCK_THREADS, 0, stream>>>(
        adj, support, b, embed1, logp, 16384);
}